// LinearAssignment_54795192762699
// MI455X (gfx1250) — compile-verified
//
#include <hip/hip_runtime.h>
#include <math.h>

// CDNA5 / gfx1250: wave32, WMMA 16x16x4 f32 used for row/col sum reductions,
// async HBM->LDS (ASYNCcnt) used for the one-shot tile load.
typedef __attribute__((ext_vector_type(2))) float v2f;
typedef __attribute__((ext_vector_type(8))) float v8f;

#define TILE 64
#define STRIDE 68          // 64 + 4 pad: keeps 16B alignment, de-correlates LDS banks
#define EPS 1e-8f
#define MAX_ITERS 100
#define LN2 0.69314718055994530942f

__device__ __forceinline__ float fast_rcp(float x) { return __builtin_amdgcn_rcpf(x); }
__device__ __forceinline__ float fast_ln(float x)  { return __builtin_amdgcn_logf(x) * LN2; }

__device__ __forceinline__ unsigned pcg_hash(unsigned v) {
  v = v * 747796405u + 2891336453u;
  unsigned w = ((v >> ((v >> 28) + 4u)) ^ v) * 277803737u;
  return (w >> 22) ^ w;
}

__global__ __launch_bounds__(256)
void sinkhorn_sample_kernel(const float* __restrict__ x,
                            float* __restrict__ out,
                            int n_items) {
  __shared__ float sP[TILE * STRIDE];   // 17 KB resident tile
  __shared__ float sPart[2][TILE];      // partial sums from the two K-halves
  __shared__ float sU[TILE];            // previous column sums (cond check)
  __shared__ float sScale[TILE];        // reciprocal scales (cols, then rows)
  __shared__ int   sFlag;

  const int t    = threadIdx.x;
  const int b    = blockIdx.x;
  const int lane = t & 31;
  const int wave = t >> 5;
  const int blk   = wave & 3;           // which 16-wide tile block
  const int khalf = wave >> 2;          // which half of the 16 K-steps
  const int n16   = lane & 15;          // M/N index inside a fragment
  const int koff  = (lane >> 4) << 1;   // K offset 0 or 2 (ISA A/B 32-bit layout)

  const float* src = x + (size_t)b * (TILE * TILE);

  // ---- async-stream the 16 KB tile straight into padded LDS (no VGPR staging) ----
  {
    #pragma unroll
    for (int i = 0; i < 4; ++i) {
      int q    = t + 256 * i;           // float4 index 0..1023
      int flat = q << 2;
      int a    = flat >> 6;
      int c    = flat & 63;
      unsigned lds_addr        = (unsigned)(size_t)&sP[a * STRIDE + c]; // LDS byte offset
      unsigned long long gaddr = (unsigned long long)(size_t)(src + flat);
      asm volatile("global_load_async_to_lds_b128 %0, %1, off"
                   :: "v"(lds_addr), "v"(gaddr) : "memory");
    }
    asm volatile("s_wait_asynccnt 0x0" ::: "memory");
  }
  if (t < TILE) sU[t] = 0.0f;
  __syncthreads();

  const v2f ones = {1.0f, 1.0f};

  for (int iter = 0; iter < MAX_ITERS; ++iter) {
    __syncthreads();                    // bar0: sP stable, prev flag-read done
    if (t == 0) sFlag = 0;

    // ---- column sums: D = ones(16x4) x P(4x16), accumulate 8 K-slabs ----
    v8f accc = {};
    const int c0 = blk << 4;
    #pragma unroll
    for (int kk = 0; kk < 8; ++kk) {
      const int r = ((khalf << 3) + kk) << 2;        // base row of 4-row K slab
      v2f bfrag;
      bfrag.x = sP[(r + koff)     * STRIDE + c0 + n16];
      bfrag.y = sP[(r + koff + 1) * STRIDE + c0 + n16];
      accc = __builtin_amdgcn_wmma_f32_16x16x4_f32(false, ones, false, bfrag,
                                                   (short)0, accc, false, false);
    }
    if (lane < 16) sPart[khalf][c0 + lane] = accc[0];   // D row M=0 == colsum
    __syncthreads();                    // bar1

    if (t < TILE) {
      float s = sPart[0][t] + sPart[1][t];
      if (fabsf(s - sU[t]) > EPS) sFlag = 1;   // benign all-write-1 race
      sU[t]     = s;
      sScale[t] = fast_rcp(s);
    }
    __syncthreads();                    // bar2
    if (sFlag == 0) break;              // cond() false for this item: stop

    // ---- normalize columns (vectorized LDS read-modify-write) ----
    {
      const int a  = t >> 2;
      const int cb = (t & 3) << 4;
      #pragma unroll
      for (int i = 0; i < 4; ++i) {
        float4 v  = *reinterpret_cast<float4*>(&sP[a * STRIDE + cb + (i << 2)]);
        float4 sc = reinterpret_cast<float4*>(sScale)[(cb >> 2) + i];
        v.x *= sc.x; v.y *= sc.y; v.z *= sc.z; v.w *= sc.w;
        *reinterpret_cast<float4*>(&sP[a * STRIDE + cb + (i << 2)]) = v;
      }
    }
    __syncthreads();                    // bar3

    // ---- row sums: D = P(16x4) x ones(4x16), accumulate 8 K-slabs ----
    v8f accr = {};
    const int r0 = blk << 4;
    #pragma unroll
    for (int kk = 0; kk < 8; ++kk) {
      const int k4 = ((khalf << 3) + kk) << 2;
      v2f afrag;
      afrag.x = sP[(r0 + n16) * STRIDE + k4 + koff];
      afrag.y = sP[(r0 + n16) * STRIDE + k4 + koff + 1];
      accr = __builtin_amdgcn_wmma_f32_16x16x4_f32(false, afrag, false, ones,
                                                   (short)0, accr, false, false);
    }
    if (lane == 0) {                    // D col N=0, M = 0..7
      #pragma unroll
      for (int v = 0; v < 8; ++v) sPart[khalf][r0 + v] = accr[v];
    } else if (lane == 16) {            // D col N=0, M = 8..15
      #pragma unroll
      for (int v = 0; v < 8; ++v) sPart[khalf][r0 + 8 + v] = accr[v];
    }
    __syncthreads();                    // bar4
    if (t < TILE) sScale[t] = fast_rcp(sPart[0][t] + sPart[1][t]);
    __syncthreads();                    // bar5

    // ---- normalize rows ----
    {
      const int a  = t >> 2;
      const int cb = (t & 3) << 4;
      const float rs = sScale[a];
      #pragma unroll
      for (int i = 0; i < 4; ++i) {
        float4 v = *reinterpret_cast<float4*>(&sP[a * STRIDE + cb + (i << 2)]);
        v.x *= rs; v.y *= rs; v.z *= rs; v.w *= rs;
        *reinterpret_cast<float4*>(&sP[a * STRIDE + cb + (i << 2)]) = v;
      }
    }
  }
  __syncthreads();

  // ---- Gumbel-max categorical sample per (item, agent); 4 lanes/agent ----
  {
    const int a  = t >> 2;
    const int j  = t & 3;
    const int cb = j << 4;
    float best = -3.4e38f, bestlp = 0.0f;
    int bestc = 0;
    #pragma unroll
    for (int i = 0; i < 16; ++i) {
      const int c = cb + i;
      float p  = sP[a * STRIDE + c];
      float lp = fast_ln(p);
      unsigned h = pcg_hash(((unsigned)b * 1664525u) ^
                            ((unsigned)a * 22695477u) ^
                            ((unsigned)c * 2654435761u) ^ 42u);
      float u = (float)(h >> 8) * 0x1.0p-24f + 0x1.0p-25f;   // (0,1)
      float g = -fast_ln(-fast_ln(u));                        // Gumbel noise
      float score = lp + g;
      if (score > best) { best = score; bestlp = lp; bestc = c; }
    }
    #pragma unroll
    for (int off = 1; off < 4; off <<= 1) {                   // wave32 shuffles
      float ob  = __shfl_xor(best,   off, 32);
      float olp = __shfl_xor(bestlp, off, 32);
      int   oc  = __shfl_xor(bestc,  off, 32);
      if (ob > best) { best = ob; bestlp = olp; bestc = oc; }
    }
    if (j == 0) {
      const size_t idx = (size_t)b * TILE + a;
      out[idx] = (float)bestc;                                // actions (f32)
      out[(size_t)n_items * TILE + idx] = bestlp;             // log_probs
    }
  }
}

extern "C" void kernel_launch(void* const* d_in, const int* in_sizes, int n_in,
                              void* d_out, int out_size, void* d_ws, size_t ws_size,
                              hipStream_t stream) {
  (void)n_in; (void)out_size; (void)d_ws; (void)ws_size;
  const float* x = (const float*)d_in[0];
  float* out = (float*)d_out;
  const int n_items = in_sizes[0] / (TILE * TILE);   // 4096
  hipLaunchKernelGGL(sinkhorn_sample_kernel,
                     dim3(n_items), dim3(256), 0, stream,
                     x, out, n_items);
}